// ProposalPool_34376918237410
// MI455X (gfx1250) — compile-verified
//
#include <hip/hip_runtime.h>
#include <cfloat>

// Triangular segment-max pooling for MI455X (gfx1250).
//   out[b,c,i,j] = max(x[b,c,i:j+1]) for i <= j, else 0
// Store-bandwidth bound: 256 MB of output vs 4 MB input (~11 us floor @23.3 TB/s).
// CDNA5 paths used: TDM tensor_load_to_lds (global->LDS DMA, with LDS row
// padding via the descriptor pad feature), s_wait_tensorcnt, and
// non-temporal global_store_b128 for the streaming 256 MB output.

typedef float        f32x4 __attribute__((ext_vector_type(4)));
typedef unsigned int u32x4 __attribute__((ext_vector_type(4)));
typedef int          i32x8 __attribute__((ext_vector_type(8)));
typedef int          i32x4 __attribute__((ext_vector_type(4)));

#define DLEN            64
#define ROWS_PER_BLOCK  16
#define LDS_STRIDE      65   // 64 data dwords + 1 pad dword inserted by TDM

__global__ __launch_bounds__(256) void
segmax_tri_kernel(const float* __restrict__ x, float* __restrict__ out, int rows) {
  __shared__ float tile[ROWS_PER_BLOCK * LDS_STRIDE];

  const int row0 = blockIdx.x * ROWS_PER_BLOCK;

  // ---- Wave 0: one TDM descriptor loads the whole block's 16x64 f32 tile ----
  if (threadIdx.x < 32) {  // wave32: this is exactly wave 0
    unsigned long long ga = (unsigned long long)(const void*)(x + (size_t)row0 * DLEN);
    unsigned int lds_addr = (unsigned int)(unsigned long long)(const void*)&tile[0];

    // D# group 0 (128b): count=1 valid | lds_addr | global_addr[56:0] | type=2
    u32x4 g0;
    g0[0] = 1u;
    g0[1] = lds_addr;
    g0[2] = (unsigned int)ga;
    g0[3] = (unsigned int)((ga >> 32) & 0x1FFFFFFu) | (2u << 30);

    // D# group 1 (256b): 2D tensor rows x 64, tile 16x64, data_size=4B,
    // pad_enable, pad_interval=5 (every 64 DWORDs), pad_amount=0 (1 DWORD)
    // -> LDS row stride becomes 65 floats (bank-conflict-free broadcasts).
    i32x8 g1;
    g1[0] = (int)((2u << 16) | (1u << 20) | (5u << 22));
    g1[1] = (int)(((unsigned)DLEN & 0xFFFFu) << 16);                 // tensor_dim0[15:0]
    g1[2] = (int)((((unsigned)DLEN >> 16) & 0xFFFFu)                 // tensor_dim0[31:16]
                | (((unsigned)rows & 0xFFFFu) << 16));               // tensor_dim1[15:0]
    g1[3] = (int)((((unsigned)rows >> 16) & 0xFFFFu)                 // tensor_dim1[31:16]
                | ((unsigned)DLEN << 16));                           // tile_dim0 = 64
    g1[4] = (int)ROWS_PER_BLOCK;                                     // tile_dim1=16, tile_dim2=0
    g1[5] = (int)DLEN;                                               // tensor_dim0_stride = 64
    g1[6] = 0;
    g1[7] = 0;

    i32x4 gz4 = {0, 0, 0, 0};              // groups 2/3 unused (<=2D tensor)
    i32x8 gz8 = {0, 0, 0, 0, 0, 0, 0, 0};  // extra group in 6-arg builtin form

    // amdgpu-toolchain (clang-23) 6-arg form:
    // (u32x4 g0, i32x8 g1, i32x4, i32x4, i32x8, i32 cpol)
    __builtin_amdgcn_tensor_load_to_lds(g0, g1, gz4, gz4, gz8, 0);
    __builtin_amdgcn_s_wait_tensorcnt(0);
  }
  __syncthreads();

  // ---- Compute: 16 threads per row, 4 consecutive output columns each ----
  const int g  = threadIdx.x >> 4;   // row group within block (0..15)
  const int t  = threadIdx.x & 15;   // column quad (0..15)
  const int r  = row0 + g;
  if (r >= rows) return;

  const int   j0    = t * 4;
  const float* lrow = &tile[g * LDS_STRIDE];
  float* obase      = out + (size_t)r * (DLEN * DLEN) + j0;

  float m0 = -FLT_MAX, m1 = -FLT_MAX, m2 = -FLT_MAX, m3 = -FLT_MAX;

  // out[i,:] = max(x[i], out[i+1,:]) on j>=i; zero on j<i.
  for (int i = DLEN - 1; i >= 0; --i) {
    const float xi = lrow[i];

    const bool b0 = (j0 + 0) >= i;
    const bool b1 = (j0 + 1) >= i;
    const bool b2 = (j0 + 2) >= i;
    const bool b3 = (j0 + 3) >= i;

    m0 = b0 ? fmaxf(m0, xi) : m0;
    m1 = b1 ? fmaxf(m1, xi) : m1;
    m2 = b2 ? fmaxf(m2, xi) : m2;
    m3 = b3 ? fmaxf(m3, xi) : m3;

    f32x4 v;
    v.x = b0 ? m0 : 0.0f;
    v.y = b1 ? m1 : 0.0f;
    v.z = b2 ? m2 : 0.0f;
    v.w = b3 ? m3 : 0.0f;

    // Streaming 16B store (output written once, never re-read; > L2 size).
    __builtin_nontemporal_store(v, (f32x4*)(obase + (size_t)i * DLEN));
  }
}

extern "C" void kernel_launch(void* const* d_in, const int* in_sizes, int n_in,
                              void* d_out, int out_size, void* d_ws, size_t ws_size,
                              hipStream_t stream) {
  const float* x  = (const float*)d_in[0];
  float*       out = (float*)d_out;
  const int rows = in_sizes[0] / DLEN;              // B*C = 16384
  const int grid = (rows + ROWS_PER_BLOCK - 1) / ROWS_PER_BLOCK;
  hipLaunchKernelGGL(segmax_tri_kernel, dim3(grid), dim3(256), 0, stream, x, out, rows);
}